// ProjGA_76287209112057
// MI455X (gfx1250) — compile-verified
//
#include <hip/hip_runtime.h>
#include <math.h>

// ---------------------------------------------------------------------------
// Problem constants (match reference setup_inputs)
// ---------------------------------------------------------------------------
static constexpr int Bn = 1024;
static constexpr int Un = 8;
static constexpr int Ln = 16;
static constexpr int Nn = 64;
static constexpr int Tn = 50;
static constexpr float EPSf = 1e-12f;

// Output layout in floats (complex64 stored as interleaved re,im):
// obj.T (B,T+1) real | wa (B,64,16) cpx | wd (B,16,8) cpx | W_A (T+1,B,64,16) | W_D (T+1,B,16,8)
static constexpr size_t OBJ_SZ    = (size_t)Bn * (Tn + 1);
static constexpr size_t WA_SZ     = (size_t)Bn * Nn * Ln * 2;
static constexpr size_t WD_SZ     = (size_t)Bn * Ln * Un * 2;
static constexpr size_t WAALL_SZ  = (size_t)(Tn + 1) * Bn * Nn * Ln * 2;
static constexpr size_t WA_OFF    = OBJ_SZ;
static constexpr size_t WD_OFF    = WA_OFF + WA_SZ;
static constexpr size_t WAALL_OFF = WD_OFF + WD_SZ;
static constexpr size_t WDALL_OFF = WAALL_OFF + WAALL_SZ;

typedef float v2f __attribute__((ext_vector_type(2)));
typedef float v8f __attribute__((ext_vector_type(8)));

__device__ inline v8f wmma4(v2f a, v2f b, v8f c) {
  // V_WMMA_F32_16X16X4_F32 : D(16x16,f32) = A(16x4,f32) * B(4x16,f32) + C
  return __builtin_amdgcn_wmma_f32_16x16x4_f32(false, a, false, b, (short)0, c,
                                               false, false);
}

// C(16x16) = A1(16xK)*B1(Kx16) + A2(16xK)*B2(Kx16); A row-major stride lda,
// B row-major stride 16, C row-major stride 16. One full wave executes this.
__device__ inline void wmma_part(const float* A1, const float* B1,
                                 const float* A2, const float* B2,
                                 int lda, int K, float* C) {
  const int lane = threadIdx.x & 31;
  const int n  = lane & 15;        // A row index m == B/C column index n
  const int kk = (lane >> 4) << 1; // lanes 16..31 carry K+2,K+3
  v8f acc = {0.f, 0.f, 0.f, 0.f, 0.f, 0.f, 0.f, 0.f};
  for (int k0 = 0; k0 < K; k0 += 4) {
    v2f a, bv;
    a.x  = A1[n * lda + k0 + kk];
    a.y  = A1[n * lda + k0 + kk + 1];
    bv.x = B1[(k0 + kk) * 16 + n];
    bv.y = B1[(k0 + kk + 1) * 16 + n];
    acc = wmma4(a, bv, acc);
  }
  for (int k0 = 0; k0 < K; k0 += 4) {
    v2f a, bv;
    a.x  = A2[n * lda + k0 + kk];
    a.y  = A2[n * lda + k0 + kk + 1];
    bv.x = B2[(k0 + kk) * 16 + n];
    bv.y = B2[(k0 + kk + 1) * 16 + n];
    acc = wmma4(a, bv, acc);
  }
  const int hh = lane >> 4;
#pragma unroll
  for (int r = 0; r < 8; ++r) C[(r + 8 * hh) * 16 + n] = acc[r];
}

// Same but A is read transposed: A[m][k] = A[(k)*lda + col0 + m].
// Output written to C rows crow0..crow0+15 (stride 16).
__device__ inline void wmma_part_AT(const float* A1, const float* B1,
                                    const float* A2, const float* B2,
                                    int lda, int col0, int K,
                                    float* C, int crow0) {
  const int lane = threadIdx.x & 31;
  const int n  = lane & 15;
  const int kk = (lane >> 4) << 1;
  v8f acc = {0.f, 0.f, 0.f, 0.f, 0.f, 0.f, 0.f, 0.f};
  for (int k0 = 0; k0 < K; k0 += 4) {
    v2f a, bv;
    a.x  = A1[(k0 + kk) * lda + col0 + n];
    a.y  = A1[(k0 + kk + 1) * lda + col0 + n];
    bv.x = B1[(k0 + kk) * 16 + n];
    bv.y = B1[(k0 + kk + 1) * 16 + n];
    acc = wmma4(a, bv, acc);
  }
  for (int k0 = 0; k0 < K; k0 += 4) {
    v2f a, bv;
    a.x  = A2[(k0 + kk) * lda + col0 + n];
    a.y  = A2[(k0 + kk + 1) * lda + col0 + n];
    bv.x = B2[(k0 + kk) * 16 + n];
    bv.y = B2[(k0 + kk + 1) * 16 + n];
    acc = wmma4(a, bv, acc);
  }
  const int hh = lane >> 4;
#pragma unroll
  for (int r = 0; r < 8; ++r) C[(crow0 + r + 8 * hh) * 16 + n] = acc[r];
}

__device__ inline float block_reduce64(float v, float* red, int tid) {
  red[tid] = v;
  __syncthreads();
  for (int s = 32; s > 0; s >>= 1) {
    if (tid < s) red[tid] += red[tid + s];
    __syncthreads();
  }
  float r = red[0];
  __syncthreads();
  return r;
}

// Cooperative Gauss-Jordan inverse of complex 8x8 HPD matrix (64 threads).
// A (row-major 8x8 re/im) destroyed; X receives inverse; logdet = sum log|pivot|.
__device__ inline void inv8_logdet(float* Are, float* Aim, float* Xre, float* Xim,
                                   float* colr, float* coli, float* logdet, int tid) {
  const int i = tid >> 3, j = tid & 7;
  Xre[tid] = (i == j) ? 1.f : 0.f;
  Xim[tid] = 0.f;
  if (tid == 0) *logdet = 0.f;
  __syncthreads();
  for (int k = 0; k < 8; ++k) {
    if (j == 0) { colr[i] = Are[i * 8 + k]; coli[i] = Aim[i * 8 + k]; }
    __syncthreads();
    float pr = colr[k], pi = coli[k];
    float d  = fmaxf(pr * pr + pi * pi, 1e-30f);
    float id = 1.f / d;
    float ipr = pr * id, ipi = -pi * id;
    if (tid == 0) *logdet += 0.5f * logf(d);
    if (i == k) {
      float ar = Are[k * 8 + j], ai = Aim[k * 8 + j];
      Are[k * 8 + j] = ar * ipr - ai * ipi;
      Aim[k * 8 + j] = ar * ipi + ai * ipr;
      float xr = Xre[k * 8 + j], xi = Xim[k * 8 + j];
      Xre[k * 8 + j] = xr * ipr - xi * ipi;
      Xim[k * 8 + j] = xr * ipi + xi * ipr;
    }
    __syncthreads();
    if (i != k) {
      float fr = colr[i], fi = coli[i];
      float ar = Are[k * 8 + j], ai = Aim[k * 8 + j];
      Are[i * 8 + j] -= fr * ar - fi * ai;
      Aim[i * 8 + j] -= fr * ai + fi * ar;
      float xr = Xre[k * 8 + j], xi = Xim[k * 8 + j];
      Xre[i * 8 + j] -= fr * xr - fi * xi;
      Xim[i * 8 + j] -= fr * xi + fi * xr;
    }
    __syncthreads();
  }
}

// ---------------------------------------------------------------------------
// Kernel 1: per-batch SVD-based wa0 construction (written into W_A[0] slot).
// Jacobi eigensolve of h h^H (8x8 Hermitian) -> top right-singular vectors,
// Householder-QR completion of the 64x64 orthonormal basis; wa0[i][j] =
// conj(Q[j][i]) / (|.|+eps).
// ---------------------------------------------------------------------------
__global__ __launch_bounds__(64) void k_wa_init(const float* __restrict__ h_re,
                                                const float* __restrict__ h_im,
                                                float* __restrict__ out) {
  const int b = blockIdx.x;
  const int tid = threadIdx.x;

  __shared__ float HRe[Un][Nn], HIm[Un][Nn];
  __shared__ float Gre[Un][Un], Gim[Un][Un];
  __shared__ float Ure[Un][Un], Uim[Un][Un];
  __shared__ float ev[Un];
  __shared__ int   ord[Un];
  __shared__ float Vre[Nn][Un], Vim[Nn][Un];
  __shared__ float Wre[Un][Nn], Wim[Un][Nn];
  __shared__ float winv[Un];
  __shared__ float sRed[64];

  float* hre = &HRe[0][0];
  float* him = &HIm[0][0];
  for (int o = tid; o < Un * Nn; o += 64) {
    hre[o] = h_re[(size_t)b * Un * Nn + o];
    him[o] = h_im[(size_t)b * Un * Nn + o];
  }
  __syncthreads();

  // G = h h^H (8x8 Hermitian)
  {
    int i = tid >> 3, j = tid & 7;
    float ar = 0.f, ai = 0.f;
    for (int k = 0; k < Nn; ++k) {
      float xr = HRe[i][k], xi = HIm[i][k];
      float yr = HRe[j][k], yi = HIm[j][k];
      ar += xr * yr + xi * yi;
      ai += xi * yr - xr * yi;
    }
    Gre[i][j] = ar; Gim[i][j] = ai;
    Ure[i][j] = (i == j) ? 1.f : 0.f; Uim[i][j] = 0.f;
  }
  __syncthreads();

  // Cyclic complex Jacobi on thread 0 (tiny 8x8; init-only cost)
  if (tid == 0) {
    for (int sw = 0; sw < 12; ++sw) {
      for (int p = 0; p < Un - 1; ++p) {
        for (int q = p + 1; q < Un; ++q) {
          float gr = Gre[p][q], gi = Gim[p][q];
          float ag = sqrtf(gr * gr + gi * gi);
          if (ag < 1e-12f) continue;
          float phr = gr / ag, phi = gi / ag;
          float tau = (Gre[q][q] - Gre[p][p]) / (2.f * ag);
          float sgn = (tau >= 0.f) ? 1.f : -1.f;
          float tt  = sgn / (fabsf(tau) + sqrtf(1.f + tau * tau));
          float c   = 1.f / sqrtf(1.f + tt * tt);
          float sr  = tt * c * phr, si = tt * c * phi;
          for (int k = 0; k < Un; ++k) { // G <- G*J (columns)
            float apr = Gre[k][p], api = Gim[k][p];
            float aqr = Gre[k][q], aqi = Gim[k][q];
            Gre[k][p] = c * apr - (sr * aqr + si * aqi);
            Gim[k][p] = c * api - (sr * aqi - si * aqr);
            Gre[k][q] = (sr * apr - si * api) + c * aqr;
            Gim[k][q] = (sr * api + si * apr) + c * aqi;
          }
          for (int k = 0; k < Un; ++k) { // G <- J^H*G (rows)
            float apr = Gre[p][k], api = Gim[p][k];
            float aqr = Gre[q][k], aqi = Gim[q][k];
            Gre[p][k] = c * apr - (sr * aqr - si * aqi);
            Gim[p][k] = c * api - (sr * aqi + si * aqr);
            Gre[q][k] = (sr * apr + si * api) + c * aqr;
            Gim[q][k] = (sr * api - si * apr) + c * aqi;
          }
          for (int k = 0; k < Un; ++k) { // U <- U*J
            float apr = Ure[k][p], api = Uim[k][p];
            float aqr = Ure[k][q], aqi = Uim[k][q];
            Ure[k][p] = c * apr - (sr * aqr + si * aqi);
            Uim[k][p] = c * api - (sr * aqi - si * aqr);
            Ure[k][q] = (sr * apr - si * api) + c * aqr;
            Uim[k][q] = (sr * api + si * apr) + c * aqi;
          }
        }
      }
    }
    for (int i2 = 0; i2 < Un; ++i2) { ev[i2] = Gre[i2][i2]; ord[i2] = i2; }
    for (int a2 = 0; a2 < Un; ++a2)
      for (int b2 = a2 + 1; b2 < Un; ++b2)
        if (ev[ord[b2]] > ev[ord[a2]]) { int t2 = ord[a2]; ord[a2] = ord[b2]; ord[b2] = t2; }
  }
  __syncthreads();

  // Vc[:,kk] = h^H u_{ord[kk]} / s  (64x8), thread tid owns row tid
  {
    int i = tid;
    for (int kk = 0; kk < Un; ++kk) {
      int k = ord[kk];
      float sk = sqrtf(fmaxf(ev[k], 0.f));
      float invs = (sk > 1e-12f) ? 1.f / sk : 0.f;
      float vr = 0.f, vi = 0.f;
      for (int u = 0; u < Un; ++u) {
        float hr = HRe[u][i], hi = -HIm[u][i]; // conj(h[u][i])
        float ur = Ure[u][k], ui = Uim[u][k];
        vr += hr * ur - hi * ui;
        vi += hr * ui + hi * ur;
      }
      Vre[i][kk] = vr * invs;
      Vim[i][kk] = vi * invs;
    }
  }
  __syncthreads();

  // Householder QR of Vc (64x8): reflectors w_k with H = I - (2/|w|^2) w w^H
  for (int k = 0; k < Un; ++k) {
    float contrib = 0.f;
    if (tid >= k) contrib = Vre[tid][k] * Vre[tid][k] + Vim[tid][k] * Vim[tid][k];
    float sig = sqrtf(block_reduce64(contrib, sRed, tid));
    float x0r = Vre[k][k], x0i = Vim[k][k];
    float ax0 = sqrtf(x0r * x0r + x0i * x0i);
    float alr, ali;
    if (ax0 > 1e-20f) { alr = -x0r / ax0 * sig; ali = -x0i / ax0 * sig; }
    else              { alr = -sig; ali = 0.f; }
    float wr_ = 0.f, wi_ = 0.f;
    if (tid >= k) {
      wr_ = Vre[tid][k]; wi_ = Vim[tid][k];
      if (tid == k) { wr_ -= alr; wi_ -= ali; }
    }
    Wre[k][tid] = wr_; Wim[k][tid] = wi_;
    float nw2 = block_reduce64(wr_ * wr_ + wi_ * wi_, sRed, tid);
    float binv = (nw2 > 1e-30f) ? 2.f / nw2 : 0.f;
    if (tid == 0) winv[k] = binv;
    for (int j = k + 1; j < Un; ++j) {
      float cr = 0.f, ci = 0.f;
      if (tid >= k) {
        cr = wr_ * Vre[tid][j] + wi_ * Vim[tid][j];
        ci = wr_ * Vim[tid][j] - wi_ * Vre[tid][j];
      }
      float dr = block_reduce64(cr, sRed, tid) * binv;
      float di = block_reduce64(ci, sRed, tid) * binv;
      if (tid >= k) {
        Vre[tid][j] -= wr_ * dr - wi_ * di;
        Vim[tid][j] -= wr_ * di + wi_ * dr;
      }
      __syncthreads();
    }
    __syncthreads();
  }
  __syncthreads();

  // Column tid of Q = H_0 ... H_7 applied to e_tid
  float qr[Nn], qi[Nn];
  for (int j2 = 0; j2 < Nn; ++j2) { qr[j2] = (j2 == tid) ? 1.f : 0.f; qi[j2] = 0.f; }
  for (int k = Un - 1; k >= 0; --k) {
    float dr = 0.f, di = 0.f;
    for (int j2 = k; j2 < Nn; ++j2) {
      float wr2 = Wre[k][j2], wi2 = Wim[k][j2];
      dr += wr2 * qr[j2] + wi2 * qi[j2];
      di += wr2 * qi[j2] - wi2 * qr[j2];
    }
    float bv = winv[k];
    dr *= bv; di *= bv;
    for (int j2 = k; j2 < Nn; ++j2) {
      float wr2 = Wre[k][j2], wi2 = Wim[k][j2];
      qr[j2] -= wr2 * dr - wi2 * di;
      qi[j2] -= wr2 * di + wi2 * dr;
    }
  }

  // wa0[tid][j] = conj(Q[j][tid]) / (|.| + eps), written to W_A[t=0]
  float2* wa0 = (float2*)(out + WAALL_OFF + (size_t)b * (Nn * Ln * 2));
  for (int j2 = 0; j2 < Ln; ++j2) {
    float re = qr[j2], im = -qi[j2];
    float inv = 1.f / (sqrtf(re * re + im * im) + EPSf);
    float2 v; v.x = re * inv; v.y = im * inv;
    wa0[tid * Ln + j2] = v;
  }
}

// ---------------------------------------------------------------------------
// Kernel 2: 50-iteration projected-gradient loop. One block (2 waves) per
// batch; hs/wa/wd resident in LDS; K=64 / M=64 complex GEMMs on the WMMA pipe.
// ---------------------------------------------------------------------------
__global__ __launch_bounds__(64) void k_iter(const float* __restrict__ h_re,
                                             const float* __restrict__ h_im,
                                             const float* __restrict__ wd0_re,
                                             const float* __restrict__ wd0_im,
                                             const float* __restrict__ mu,
                                             const int* __restrict__ noise_var,
                                             float* __restrict__ out) {
  const int b   = blockIdx.x;
  const int tid = threadIdx.x;
  const int wv  = tid >> 5;

  __shared__ float sHSre[16 * 64], sHSim[16 * 64], sHSimN[16 * 64];
  __shared__ float sWAre[64 * 16], sWAim[64 * 16];
  __shared__ float sGWre[64 * 16], sGWim[64 * 16];
  __shared__ float sHWAre[16 * 16], sHWAim[16 * 16];
  __shared__ float sWDre[16 * 8], sWDim[16 * 8];
  __shared__ float sWDtr[16 * 8], sWDti[16 * 8];
  __shared__ float sHwre[64], sHwim[64];
  __shared__ float sMre[64], sMim[64];
  __shared__ float sXre[64], sXim[64];
  __shared__ float sPre[64], sPim[64];
  __shared__ float sG8re[8 * 16], sG8im[8 * 16];
  __shared__ float sQre[16 * 8], sQim[16 * 8];
  __shared__ float sRre[16 * 16], sRim[16 * 16];
  __shared__ float sGDre[16 * 8], sGDim[16 * 8];
  __shared__ float sColr[8], sColi[8];
  __shared__ float sRed[64];
  __shared__ float sLogdet;

  const float scale = sqrtf(1.f / ((float)Un * (float)noise_var[0]));

  // hs padded to 16 rows (rows 8..15 zero); keep -im copy for WMMA subtraction
  for (int o = tid; o < 16 * 64; o += 64) {
    float re = 0.f, im = 0.f;
    if ((o >> 6) < Un) {
      re = scale * h_re[(size_t)b * Un * Nn + o];
      im = scale * h_im[(size_t)b * Un * Nn + o];
    }
    sHSre[o] = re; sHSim[o] = im; sHSimN[o] = -im;
  }
  { // wa0 from W_A[0] (written by k_wa_init earlier on this stream)
    const float2* wa0 = (const float2*)(out + WAALL_OFF + (size_t)b * (Nn * Ln * 2));
    for (int o = tid; o < Nn * Ln; o += 64) {
      float2 v = wa0[o];
      sWAre[o] = v.x; sWAim[o] = v.y;
    }
  }
  for (int o = tid; o < Ln * Un; o += 64) {
    sWDtr[o] = wd0_re[(size_t)b * Ln * Un + o];
    sWDti[o] = wd0_im[(size_t)b * Ln * Un + o];
  }
  __syncthreads();

  auto normalize_wd = [&]() { // wd = sqrt(U/||wa@wdt||_F^2) * wdt
    float part = 0.f;
    for (int j = 0; j < Un; ++j) {
      float ar = 0.f, ai = 0.f;
      for (int k = 0; k < Ln; ++k) {
        float xr = sWAre[tid * Ln + k], xi = sWAim[tid * Ln + k];
        float yr = sWDtr[k * Un + j],   yi = sWDti[k * Un + j];
        ar += xr * yr - xi * yi;
        ai += xr * yi + xi * yr;
      }
      part += ar * ar + ai * ai;
    }
    float f2 = block_reduce64(part, sRed, tid);
    float gam = sqrtf((float)Un / fmaxf(f2, 1e-30f));
    for (int o = tid; o < Ln * Un; o += 64) {
      sWDre[o] = gam * sWDtr[o];
      sWDim[o] = gam * sWDti[o];
    }
    __syncthreads();
  };

  auto compute_hwa = [&]() { // Hwa = hs @ wa  (complex 16x16, K=64, WMMA)
    if (wv == 0) wmma_part(sHSre, sWAre, sHSimN, sWAim, 64, 64, sHWAre);
    else         wmma_part(sHSre, sWAim, sHSim,  sWAre, 64, 64, sHWAim);
    __syncthreads();
  };

  auto m_inv_obj = [&]() { // Hw = Hwa@wd ; M = I + Hw Hw^H ; X = M^-1 ; logdet
    {
      int i = tid >> 3, j = tid & 7;
      float ar = 0.f, ai = 0.f;
      for (int k = 0; k < Ln; ++k) {
        float xr = sHWAre[i * 16 + k], xi = sHWAim[i * 16 + k];
        float yr = sWDre[k * Un + j],  yi = sWDim[k * Un + j];
        ar += xr * yr - xi * yi;
        ai += xr * yi + xi * yr;
      }
      sHwre[tid] = ar; sHwim[tid] = ai;
    }
    __syncthreads();
    {
      int i = tid >> 3, j = tid & 7;
      float ar = (i == j) ? 1.f : 0.f, ai = 0.f;
      for (int k = 0; k < Un; ++k) {
        float xr = sHwre[i * 8 + k], xi = sHwim[i * 8 + k];
        float yr = sHwre[j * 8 + k], yi = -sHwim[j * 8 + k];
        ar += xr * yr - xi * yi;
        ai += xr * yi + xi * yr;
      }
      sMre[tid] = ar; sMim[tid] = ai;
    }
    __syncthreads();
    inv8_logdet(sMre, sMim, sXre, sXim, sColr, sColi, &sLogdet, tid);
  };

  // ---- t = 0 state & outputs ----
  normalize_wd();
  compute_hwa();
  m_inv_obj();
  {
    float2* wdo = (float2*)(out + WDALL_OFF + (size_t)b * (Ln * Un * 2));
    for (int o = tid; o < Ln * Un; o += 64) {
      float2 v; v.x = sWDre[o]; v.y = sWDim[o]; wdo[o] = v;
    }
    if (tid == 0) out[(size_t)b * (Tn + 1)] = fabsf(sLogdet);
  }

  // ---- iterations ----
  for (int it = 0; it < Tn; ++it) {
    const float mu0 = mu[2 * it], mu1 = mu[2 * it + 1];

    // P = X^T conj(Hw)   (8x8)
    {
      int i = tid >> 3, j = tid & 7;
      float pr = 0.f, pi = 0.f;
      for (int k = 0; k < Un; ++k) {
        float xr = sXre[k * 8 + i], xi = sXim[k * 8 + i];
        float hr = sHwre[k * 8 + j], hi = -sHwim[k * 8 + j];
        pr += xr * hr - xi * hi;
        pi += xr * hi + xi * hr;
      }
      sPre[tid] = pr; sPim[tid] = pi;
    }
    __syncthreads();
    // G8 = P @ wd^T   (8x16)
    for (int o = tid; o < 8 * 16; o += 64) {
      int i = o >> 4, j = o & 15;
      float ar = 0.f, ai = 0.f;
      for (int k = 0; k < Un; ++k) {
        float xr = sPre[i * 8 + k], xi = sPim[i * 8 + k];
        float yr = sWDre[j * Un + k], yi = sWDim[j * Un + k];
        ar += xr * yr - xi * yi;
        ai += xr * yi + xi * yr;
      }
      sG8re[o] = ar; sG8im[o] = ai;
    }
    __syncthreads();
    // grad_wa = hs^T @ G8  (64x16, WMMA; wave w handles tiles {w, w+2})
    for (int tile = wv; tile < 4; tile += 2) {
      wmma_part_AT(sHSre, sG8re, sHSimN, sG8im, 64, tile * 16, 8, sGWre, tile * 16);
      wmma_part_AT(sHSre, sG8im, sHSim,  sG8re, 64, tile * 16, 8, sGWim, tile * 16);
    }
    __syncthreads();
    // wa = (wa + mu0*grad) / |wa_t + eps|
    for (int o = tid; o < Nn * Ln; o += 64) {
      float wr = sWAre[o] + mu0 * sGWre[o];
      float wi = sWAim[o] + mu0 * sGWim[o];
      float dr = wr + EPSf;
      float inv = 1.f / fmaxf(sqrtf(dr * dr + wi * wi), 1e-30f);
      sWAre[o] = wr * inv;
      sWAim[o] = wi * inv;
    }
    __syncthreads();

    compute_hwa();   // Hwa with wa_new
    m_inv_obj();     // X = inv(M(wa_new, wd_old))

    // Q = Hwa^T @ X^T  (16x8)
    for (int o = tid; o < Ln * Un; o += 64) {
      int i = o >> 3, j = o & 7;
      float ar = 0.f, ai = 0.f;
      for (int k = 0; k < Un; ++k) {
        float xr = sHWAre[k * 16 + i], xi = sHWAim[k * 16 + i];
        float yr = sXre[j * 8 + k],    yi = sXim[j * 8 + k];
        ar += xr * yr - xi * yi;
        ai += xr * yi + xi * yr;
      }
      sQre[o] = ar; sQim[o] = ai;
    }
    __syncthreads();
    // R = Q @ conj(Hwa)  (16x16)
    for (int o = tid; o < 16 * 16; o += 64) {
      int i = o >> 4, j = o & 15;
      float ar = 0.f, ai = 0.f;
      for (int k = 0; k < Un; ++k) {
        float xr = sQre[i * 8 + k], xi = sQim[i * 8 + k];
        float yr = sHWAre[k * 16 + j], yi = -sHWAim[k * 16 + j];
        ar += xr * yr - xi * yi;
        ai += xr * yi + xi * yr;
      }
      sRre[o] = ar; sRim[o] = ai;
    }
    __syncthreads();
    // grad_wd = R @ conj(wd)  (16x8)
    for (int o = tid; o < Ln * Un; o += 64) {
      int i = o >> 3, j = o & 7;
      float ar = 0.f, ai = 0.f;
      for (int k = 0; k < Ln; ++k) {
        float xr = sRre[i * 16 + k], xi = sRim[i * 16 + k];
        float yr = sWDre[k * Un + j], yi = -sWDim[k * Un + j];
        ar += xr * yr - xi * yi;
        ai += xr * yi + xi * yr;
      }
      sGDre[o] = ar; sGDim[o] = ai;
    }
    __syncthreads();
    for (int o = tid; o < Ln * Un; o += 64) {
      sWDtr[o] = sWDre[o] + mu1 * sGDre[o];
      sWDti[o] = sWDim[o] + mu1 * sGDim[o];
    }
    __syncthreads();
    normalize_wd();
    m_inv_obj();     // obj(wa_new, wd_new); X/Hw carried into next iteration

    // outputs for step it+1 (coalesced float2 stores)
    {
      float2* wao = (float2*)(out + WAALL_OFF +
                              ((size_t)(it + 1) * Bn + b) * (Nn * Ln * 2));
      for (int o = tid; o < Nn * Ln; o += 64) {
        float2 v; v.x = sWAre[o]; v.y = sWAim[o]; wao[o] = v;
      }
      float2* wdo = (float2*)(out + WDALL_OFF +
                              ((size_t)(it + 1) * Bn + b) * (Ln * Un * 2));
      for (int o = tid; o < Ln * Un; o += 64) {
        float2 v; v.x = sWDre[o]; v.y = sWDim[o]; wdo[o] = v;
      }
      if (tid == 0) out[(size_t)b * (Tn + 1) + (it + 1)] = fabsf(sLogdet);
    }
  }

  // final wa/wd slots
  {
    float2* wao = (float2*)(out + WA_OFF + (size_t)b * (Nn * Ln * 2));
    for (int o = tid; o < Nn * Ln; o += 64) {
      float2 v; v.x = sWAre[o]; v.y = sWAim[o]; wao[o] = v;
    }
    float2* wdo = (float2*)(out + WD_OFF + (size_t)b * (Ln * Un * 2));
    for (int o = tid; o < Ln * Un; o += 64) {
      float2 v; v.x = sWDre[o]; v.y = sWDim[o]; wdo[o] = v;
    }
  }
}

extern "C" void kernel_launch(void* const* d_in, const int* in_sizes, int n_in,
                              void* d_out, int out_size, void* d_ws, size_t ws_size,
                              hipStream_t stream) {
  (void)in_sizes; (void)n_in; (void)out_size; (void)d_ws; (void)ws_size;
  const float* h_re   = (const float*)d_in[0];
  const float* h_im   = (const float*)d_in[1];
  const float* wd0_re = (const float*)d_in[2];
  const float* wd0_im = (const float*)d_in[3];
  const float* mu     = (const float*)d_in[4];
  const int*   noise  = (const int*)d_in[8]; // noise_var_DL
  float* out = (float*)d_out;

  k_wa_init<<<Bn, 64, 0, stream>>>(h_re, h_im, out);
  k_iter<<<Bn, 64, 0, stream>>>(h_re, h_im, wd0_re, wd0_im, mu, noise, out);
}